// GraphAttentionV2Layer_10548439679700
// MI455X (gfx1250) — compile-verified
//
#include <hip/hip_runtime.h>

typedef __attribute__((ext_vector_type(2))) float v2f;
typedef __attribute__((ext_vector_type(8))) float v8f;

#define N_NODES 1024
#define IN_FEAT 256
#define OUT_FEAT 128
#define NHEAD 4
#define HF 32
#define NEG_SLOPE 0.2f

// ---------------------------------------------------------------------------
// Kernel 1: g_l = h @ Wl^T, g_r = h @ Wr^T  (f32 WMMA 16x16x4)
// One wave (32 threads) computes one 16x16 tile of g. blockIdx.z picks Wl/Wr.
// A-matrix (16x4, M=node, K=in-feature): lane m=lane&15, K-pair = (lane>>4)*2.
// B-matrix (4x16, K=in-feature, N=out-feature): B[k][n] = W[n][k] -> contiguous
// float2 along k in row-major W. C layout: lane<16 -> M=v, lane>=16 -> M=v+8.
// ---------------------------------------------------------------------------
__global__ __launch_bounds__(32) void gat_gemm_hW(
    const float* __restrict__ hmat,
    const float* __restrict__ Wl,
    const float* __restrict__ Wr,
    float* __restrict__ gl,
    float* __restrict__ gr) {
  const int lane = threadIdx.x;        // 0..31
  const int m    = lane & 15;          // row within tile (A) / col within tile (B)
  const int kb   = (lane >> 4) << 1;   // 0 or 2: which K pair this half-wave owns
  const int i0   = blockIdx.x * 16;    // node tile base
  const int o0   = blockIdx.y * 16;    // out-feature tile base
  const float* W = blockIdx.z ? Wr : Wl;
  float*       g = blockIdx.z ? gr : gl;

  const float* arow = hmat + (size_t)(i0 + m) * IN_FEAT + kb;  // h[i0+m][kb + k]
  const float* brow = W    + (size_t)(o0 + m) * IN_FEAT + kb;  // W[o0+n][kb + k]

  v8f c = {};
  #pragma unroll 4
  for (int k = 0; k < IN_FEAT; k += 4) {
    v2f a = *reinterpret_cast<const v2f*>(arow + k);
    v2f b = *reinterpret_cast<const v2f*>(brow + k);
    c = __builtin_amdgcn_wmma_f32_16x16x4_f32(false, a, false, b, (short)0, c,
                                              false, false);
  }

  const int mbase = (lane < 16) ? 0 : 8;
  #pragma unroll
  for (int v = 0; v < 8; ++v) {
    g[(size_t)(i0 + mbase + v) * OUT_FEAT + o0 + m] = c[v];
  }
}

// ---------------------------------------------------------------------------
// Kernel 2: masked attention logits + softmax over source nodes j.
// One block per destination node i; 256 threads each own 4 j values.
// e[i,j,h] = sum_f attn_w[f] * leakyrelu(g_l[j,h,f] + g_r[i,h,f]); adj mask;
// softmax over j per head via LDS tree reductions. Writes a[h][i][j] (f32).
// ---------------------------------------------------------------------------
__global__ __launch_bounds__(256) void gat_attn_softmax(
    const float* __restrict__ gl,
    const float* __restrict__ gr,
    const int* __restrict__ adj,
    const float* __restrict__ attn_w,
    float* __restrict__ amat) {
  const int i = blockIdx.x;
  const int t = threadIdx.x;

  __shared__ float sr[OUT_FEAT];   // g_r row for node i
  __shared__ float sw[HF];         // attn_w
  __shared__ float red[NHEAD][256];

  if (t < OUT_FEAT) sr[t] = gr[(size_t)i * OUT_FEAT + t];
  if (t < HF)       sw[t] = attn_w[t];
  __syncthreads();

  float ev[4][NHEAD];
  #pragma unroll
  for (int jj = 0; jj < 4; ++jj) {
    const int j = t + jj * 256;
    const int msk = adj[(size_t)i * N_NODES + j];
    const float* glr = gl + (size_t)j * OUT_FEAT;
    #pragma unroll
    for (int hh = 0; hh < NHEAD; ++hh) {
      float acc = 0.0f;
      #pragma unroll
      for (int f = 0; f < HF; ++f) {
        float s = glr[hh * HF + f] + sr[hh * HF + f];
        float act = (s >= 0.0f) ? s : NEG_SLOPE * s;
        acc = fmaf(sw[f], act, acc);
      }
      ev[jj][hh] = msk ? acc : -__builtin_inff();
    }
  }

  // --- block-wide max per head ---
  #pragma unroll
  for (int hh = 0; hh < NHEAD; ++hh) {
    float mloc = ev[0][hh];
    #pragma unroll
    for (int jj = 1; jj < 4; ++jj) mloc = fmaxf(mloc, ev[jj][hh]);
    red[hh][t] = mloc;
  }
  __syncthreads();
  for (int s = 128; s > 0; s >>= 1) {
    if (t < s) {
      #pragma unroll
      for (int hh = 0; hh < NHEAD; ++hh)
        red[hh][t] = fmaxf(red[hh][t], red[hh][t + s]);
    }
    __syncthreads();
  }
  float mx[NHEAD];
  #pragma unroll
  for (int hh = 0; hh < NHEAD; ++hh) mx[hh] = red[hh][0];
  __syncthreads();

  // --- exp + block-wide sum per head ---
  #pragma unroll
  for (int hh = 0; hh < NHEAD; ++hh) {
    float sloc = 0.0f;
    #pragma unroll
    for (int jj = 0; jj < 4; ++jj) {
      ev[jj][hh] = expf(ev[jj][hh] - mx[hh]);
      sloc += ev[jj][hh];
    }
    red[hh][t] = sloc;
  }
  __syncthreads();
  for (int s = 128; s > 0; s >>= 1) {
    if (t < s) {
      #pragma unroll
      for (int hh = 0; hh < NHEAD; ++hh)
        red[hh][t] += red[hh][t + s];
    }
    __syncthreads();
  }
  float inv[NHEAD];
  #pragma unroll
  for (int hh = 0; hh < NHEAD; ++hh) inv[hh] = 1.0f / red[hh][0];

  // --- write normalized attention: a[h][i][j] ---
  #pragma unroll
  for (int jj = 0; jj < 4; ++jj) {
    const int j = t + jj * 256;
    #pragma unroll
    for (int hh = 0; hh < NHEAD; ++hh) {
      amat[(size_t)hh * N_NODES * N_NODES + (size_t)i * N_NODES + j] =
          ev[jj][hh] * inv[hh];
    }
  }
}

// ---------------------------------------------------------------------------
// Kernel 3: h'[i, h*32+f] = ELU( sum_j a[h][i][j] * g_r[j, h*32+f] )
// Per head: C[1024x32] = A[1024x1024] @ B[1024x32] via f32 WMMA 16x16x4.
// Grid: (M tiles = 64, N tiles = 2 per head, heads = 4), one wave per tile.
// ---------------------------------------------------------------------------
__global__ __launch_bounds__(32) void gat_aggregate(
    const float* __restrict__ amat,
    const float* __restrict__ gr,
    float* __restrict__ out) {
  const int lane = threadIdx.x;
  const int m    = lane & 15;
  const int kb   = (lane >> 4) << 1;
  const int i0   = blockIdx.x * 16;      // destination-node tile
  const int f0   = blockIdx.y * 16;      // feature tile within the head (0/16)
  const int hh   = blockIdx.z;           // head

  const float* A    = amat + (size_t)hh * N_NODES * N_NODES;
  const float* arow = A + (size_t)(i0 + m) * N_NODES + kb;
  const float* bcol = gr + hh * HF + f0 + m;  // column n = m, stride OUT_FEAT in k

  v8f c = {};
  #pragma unroll 4
  for (int k = 0; k < N_NODES; k += 4) {
    v2f av = *reinterpret_cast<const v2f*>(arow + k);
    v2f bv = { bcol[(size_t)(k + kb) * OUT_FEAT],
               bcol[(size_t)(k + kb + 1) * OUT_FEAT] };
    c = __builtin_amdgcn_wmma_f32_16x16x4_f32(false, av, false, bv, (short)0, c,
                                              false, false);
  }

  const int mbase = (lane < 16) ? 0 : 8;
  #pragma unroll
  for (int v = 0; v < 8; ++v) {
    float x = c[v];
    float y = (x > 0.0f) ? x : expm1f(x);   // ELU(alpha=1)
    out[(size_t)(i0 + mbase + v) * OUT_FEAT + hh * HF + f0 + m] = y;
  }
}

// ---------------------------------------------------------------------------
extern "C" void kernel_launch(void* const* d_in, const int* in_sizes, int n_in,
                              void* d_out, int out_size, void* d_ws,
                              size_t ws_size, hipStream_t stream) {
  const float* hmat   = (const float*)d_in[0];  // [1024,256]
  const int*   adj    = (const int*)d_in[1];    // [1024,1024,1]
  const float* Wl     = (const float*)d_in[2];  // [128,256]
  const float* Wr     = (const float*)d_in[3];  // [128,256]
  const float* attn_w = (const float*)d_in[4];  // [32]
  float*       out    = (float*)d_out;          // [1024,128]

  // Workspace layout: g_l (512KB) | g_r (512KB) | a (16MB)
  float* gl = (float*)d_ws;
  float* gr = gl + (size_t)N_NODES * OUT_FEAT;
  float* a  = gr + (size_t)N_NODES * OUT_FEAT;

  gat_gemm_hW<<<dim3(N_NODES / 16, OUT_FEAT / 16, 2), 32, 0, stream>>>(
      hmat, Wl, Wr, gl, gr);
  gat_attn_softmax<<<dim3(N_NODES), 256, 0, stream>>>(gl, gr, adj, attn_w, a);
  gat_aggregate<<<dim3(N_NODES / 16, 2, NHEAD), 32, 0, stream>>>(a, gr, out);
}